// RelPositionMultiheadAttentionWeights_43722767073786
// MI455X (gfx1250) — compile-verified
//
#include <hip/hip_runtime.h>
#include <hip/hip_bf16.h>

// ---------------------------------------------------------------------------
// RelPositionMultiheadAttentionWeights for MI455X (gfx1250, wave32, WMMA)
//
// out = [attn_weights (8,2048,2304) f32 | new_cached (256,256) f32]
//
// Workspace layout (f32):
//   proj : [2048][544]      (q | k_new | p)
//   pe   : [4351][8][4]     (pos_emb @ linear_pos_w^T, head-major float4)
// ---------------------------------------------------------------------------

typedef __attribute__((ext_vector_type(2))) float v2f;
typedef __attribute__((ext_vector_type(8))) float v8f;

#define EMBED   512
#define PROJD   544
#define SEQ     2048
#define S2LEN   2304
#define LCTX    256
#define NH      8
#define DQ      32
#define POSL    4351
#define POSD    192
#define SCSTR   2308   // padded LDS row stride (floats)

__device__ __forceinline__ v8f wmma_f32_16x16x4(v2f a, v2f b, v8f c) {
    // D = A(16x4,f32) * B(4x16,f32) + C(16x16,f32)
    return __builtin_amdgcn_wmma_f32_16x16x4_f32(
        /*neg_a=*/false, a, /*neg_b=*/false, b,
        /*c_mod=*/(short)0, c, /*reuse_a=*/false, /*reuse_b=*/false);
}

// ---------------------------------------------------------------------------
// Kernel 1: proj[s][n] = sum_k x[s][k] * W[n][k] + bias[n]
// 128 (M) x 34 (N) = 4352 16x16 tiles; 8 waves/block -> 544 blocks.
// ---------------------------------------------------------------------------
__global__ void __launch_bounds__(256)
proj_gemm_kernel(const float* __restrict__ x, const float* __restrict__ W,
                 const float* __restrict__ bias, float* __restrict__ proj) {
    const int wave = threadIdx.x >> 5;
    const int lane = threadIdx.x & 31;
    const int half = lane >> 4;       // 0: lanes 0-15, 1: lanes 16-31
    const int lm   = lane & 15;
    const int tile = blockIdx.x * 8 + wave;       // 0..4351
    const int mt = tile / 34, nt = tile % 34;
    const int s0 = mt * 16, n0 = nt * 16;

    const float* arow = x + (size_t)(s0 + lm) * EMBED;  // lane's A row (M = lm)
    const float* brow = W + (size_t)(n0 + lm) * EMBED;  // lane's B col (N = lm)

    v8f c = {};
#pragma unroll 4
    for (int kc = 0; kc < EMBED / 4; ++kc) {
        const float4 fa = *(const float4*)(arow + 4 * kc);
        const float4 fb = *(const float4*)(brow + 4 * kc);
        v2f a, b;
        a.x = half ? fa.z : fa.x;  a.y = half ? fa.w : fa.y;   // K = {4kc,4kc+1} | {4kc+2,4kc+3}
        b.x = half ? fb.z : fb.x;  b.y = half ? fb.w : fb.y;
        c = wmma_f32_16x16x4(a, b, c);
    }
    const float bb = bias[n0 + lm];
#pragma unroll
    for (int j = 0; j < 8; ++j) {   // C layout: lane holds col N=lm, rows M=j+8*half
        proj[(size_t)(s0 + j + 8 * half) * PROJD + n0 + lm] = c[j] + bb;
    }
}

// ---------------------------------------------------------------------------
// Kernel 2: pe[j][o] = sum_c pos_emb[j][c] * lpw[o][c]   (o = h*4+e, 32 outputs)
// 4 waves/block, wave handles one j row, lane = output column.
// ---------------------------------------------------------------------------
__global__ void __launch_bounds__(128)
pe_gemm_kernel(const float* __restrict__ pos_emb, const float* __restrict__ lpw,
               float* __restrict__ pe) {
    const int wave = threadIdx.x >> 5;
    const int lane = threadIdx.x & 31;
    const int j = blockIdx.x * 4 + wave;
    if (j >= POSL) return;
    const float* prow = pos_emb + (size_t)j * POSD;
    const float* wrow = lpw + (size_t)lane * POSD;
    float acc = 0.f;
#pragma unroll 4
    for (int c = 0; c < POSD; c += 4) {
        const float4 p4 = *(const float4*)(prow + c);
        const float4 w4 = *(const float4*)(wrow + c);
        acc = fmaf(p4.x, w4.x, acc);
        acc = fmaf(p4.y, w4.y, acc);
        acc = fmaf(p4.z, w4.z, acc);
        acc = fmaf(p4.w, w4.w, acc);
    }
    pe[(size_t)j * 32 + lane] = acc;
}

// ---------------------------------------------------------------------------
// Kernel 3: fused attention scores + rel-pos + mask + softmax.
// grid = 8 heads * 128 s-blocks; block = 512 threads (16 waves).
// Each wave computes 9 of 144 column tiles (16x16 each) via f32 WMMA,
// adds pos term pos[s][t] = dot4(p[s,h], pe[t+2047-s, h]), masks, stores to
// LDS strip [16][2304], then wave w softmaxes row w and writes output.
// ---------------------------------------------------------------------------
__global__ void __launch_bounds__(512)
attn_softmax_kernel(const float* __restrict__ proj,
                    const float* __restrict__ lck,
                    const float* __restrict__ pe,
                    const unsigned char* __restrict__ kpm,
                    float* __restrict__ out) {
    extern __shared__ float sc[];   // [16][SCSTR]

    const int h  = blockIdx.x >> 7;
    const int s0 = (blockIdx.x & 127) * 16;
    const int wave = threadIdx.x >> 5;
    const int lane = threadIdx.x & 31;
    const int half = lane >> 4;
    const int lm   = lane & 15;

    // ---- A fragments: q rows s0..s0+15, head h (16x32 f32, K=32 -> 8 frags)
    const float* qrow = proj + (size_t)(s0 + lm) * PROJD + h * DQ;
    v2f afrag[8];
#pragma unroll
    for (int kc = 0; kc < 8; ++kc) {
        const float4 f = *(const float4*)(qrow + 4 * kc);
        afrag[kc].x = half ? f.z : f.x;
        afrag[kc].y = half ? f.w : f.y;
    }
    // ---- p rows for this lane's 8 C rows (M = j + 8*half)
    float4 prow[8];
#pragma unroll
    for (int j = 0; j < 8; ++j) {
        prow[j] = *(const float4*)(proj + (size_t)(s0 + j + 8 * half) * PROJD
                                   + 512 + h * 4);
    }

    // ---- 144 column tiles distributed over 16 waves (9 each, no divergence)
    for (int tt = wave; tt < 144; tt += 16) {
        const int t0   = tt * 16;
        const int tcol = t0 + lm;                 // this lane's key row (B col)
        const float* krow = (tcol < LCTX)
            ? (lck  + (size_t)tcol * 256 + h * DQ)
            : (proj + (size_t)(tcol - LCTX) * PROJD + 256 + h * DQ);

        if (tt + 16 < 144) {                      // uniform branch: EXEC stays full
            const int tn = tcol + 256;
            const float* nk = (tn < LCTX)
                ? (lck  + (size_t)tn * 256 + h * DQ)
                : (proj + (size_t)(tn - LCTX) * PROJD + 256 + h * DQ);
            __builtin_prefetch(nk, 0, 1);         // global_prefetch_b8
        }

        v8f c = {};
#pragma unroll
        for (int kc = 0; kc < 8; ++kc) {
            const float4 f = *(const float4*)(krow + 4 * kc);
            v2f b;
            b.x = half ? f.z : f.x;
            b.y = half ? f.w : f.y;
            c = wmma_f32_16x16x4(afrag[kc], b, c);
        }

        const bool masked = (kpm[tcol] != 0);
#pragma unroll
        for (int j = 0; j < 8; ++j) {
            const int s = s0 + j + 8 * half;
            const int pj = tcol + 2047 - s;       // rel-shift index, 0..4350
            const float4 pv = *(const float4*)(pe + (size_t)pj * 32 + h * 4);
            float v = c[j];
            v = fmaf(prow[j].x, pv.x, v);
            v = fmaf(prow[j].y, pv.y, v);
            v = fmaf(prow[j].z, pv.z, v);
            v = fmaf(prow[j].w, pv.w, v);
            if (masked) v = -1000.0f;
            sc[(j + 8 * half) * SCSTR + tcol] = v;
        }
    }
    __syncthreads();

    // ---- softmax: wave w owns row w (2304 = 72 * 32, no tail)
    float* row = sc + wave * SCSTR;
    float mx = -3.4e38f;
    for (int t = lane; t < S2LEN; t += 32) mx = fmaxf(mx, row[t]);
#pragma unroll
    for (int off = 16; off; off >>= 1) mx = fmaxf(mx, __shfl_xor(mx, off, 32));

    float sum = 0.f;
    for (int t = lane; t < S2LEN; t += 32) {
        const float e = __expf(row[t] - mx);
        row[t] = e;
        sum += e;
    }
#pragma unroll
    for (int off = 16; off; off >>= 1) sum += __shfl_xor(sum, off, 32);
    const float inv = 1.0f / sum;

    float* obase = out + ((size_t)h * SEQ + (s0 + wave)) * S2LEN;
    for (int t = lane; t < S2LEN; t += 32) obase[t] = row[t] * inv;
}

// ---------------------------------------------------------------------------
// Kernel 4: new_cached[i][c] = k_cat[2040+i][c] = proj[1784+i][256+c]
// ---------------------------------------------------------------------------
__global__ void __launch_bounds__(256)
cache_copy_kernel(const float* __restrict__ proj, float* __restrict__ out2) {
    const int i = blockIdx.x * 256 + threadIdx.x;   // 0..65535
    const int r = i >> 8, c = i & 255;
    out2[i] = proj[(size_t)(1784 + r) * PROJD + 256 + c];
}

// ---------------------------------------------------------------------------
extern "C" void kernel_launch(void* const* d_in, const int* in_sizes, int n_in,
                              void* d_out, int out_size, void* d_ws, size_t ws_size,
                              hipStream_t stream) {
    const float* x        = (const float*)d_in[0];          // [2048,512]
    const float* pos_emb  = (const float*)d_in[1];          // [4351,192]
    const float* lck      = (const float*)d_in[2];          // [256,256]
    const unsigned char* kpm = (const unsigned char*)d_in[3]; // [2304] bool
    const float* in_w     = (const float*)d_in[4];          // [544,512]
    const float* in_b     = (const float*)d_in[5];          // [544]
    const float* lpw      = (const float*)d_in[6];          // [32,192]

    float* proj = (float*)d_ws;                             // 2048*544
    float* pe   = proj + (size_t)SEQ * PROJD;               // 4351*32

    float* attn_out  = (float*)d_out;                       // 8*2048*2304
    float* cache_out = attn_out + (size_t)NH * SEQ * S2LEN; // 256*256

    proj_gemm_kernel<<<dim3(544), dim3(256), 0, stream>>>(x, in_w, in_b, proj);
    pe_gemm_kernel<<<dim3((POSL + 3) / 4), dim3(128), 0, stream>>>(pos_emb, lpw, pe);

    const size_t smem = (size_t)16 * SCSTR * sizeof(float); // 147712 B
    attn_softmax_kernel<<<dim3(NH * 128), dim3(512), smem, stream>>>(
        proj, lck, pe, kpm, attn_out);

    cache_copy_kernel<<<dim3(256), dim3(256), 0, stream>>>(proj, cache_out);
}